// MFVISemanticRoleLabeling_33904471834782
// MI455X (gfx1250) — compile-verified
//
#include <hip/hip_runtime.h>
#include <stdint.h>

// CDNA5 / gfx1250, wave32. Uses V_WMMA_F32_16X16X4_F32 (native f32 WMMA).
typedef __attribute__((ext_vector_type(2))) float v2f;
typedef __attribute__((ext_vector_type(8))) float v8f;

#define Bv   4
#define Sv   64
#define BH   (Bv * Sv)        // 256 (b,h) pairs
#define BSS  (Bv * Sv * Sv)   // 16384 elements per (B,S,S) tensor

// ---------------------------------------------------------------------------
// Kernel 1: q = sigmoid(base), plus transposed copies qT[b,m,k] = q[b,k,m]
// (for the 'bhmk,bkm->bhm' cop terms -> contiguous-k reads in the update).
// ---------------------------------------------------------------------------
__global__ void mfvi_sigmoid_kernel(const float* __restrict__ base_b,
                                    const float* __restrict__ base_e,
                                    float* __restrict__ qb,
                                    float* __restrict__ qe,
                                    float* __restrict__ qTb,
                                    float* __restrict__ qTe) {
    int idx = blockIdx.x * blockDim.x + threadIdx.x;
    if (idx >= BSS) return;
    int b = idx >> 12;
    int r = idx & 4095;
    int i = r >> 6;           // k
    int j = r & 63;           // m
    float sb = 1.0f / (1.0f + __expf(-base_b[idx]));
    float se = 1.0f / (1.0f + __expf(-base_e[idx]));
    qb[idx] = sb;
    qe[idx] = se;
    int tidx = (b << 12) + (j << 6) + i;
    qTb[tidx] = sb;
    qTe[tidx] = se;
}

// ---------------------------------------------------------------------------
// Kernel 2: one MFVI iteration.
// grid = 256 blocks (one per (b,h)), block = 256 threads (8 waves).
// Wave w: m-tile = w&3, k-half = w>>2 (k in [32*kh, 32*kh+32)).
// Each wave runs 8 k-steps of K=4, 6 WMMAs/step into two f32 accumulators.
// Column 0 of each 16x16 D tile is the matvec; the two k-half partials are
// combined through LDS (fixed order -> bitwise deterministic).
// ---------------------------------------------------------------------------
__global__ __launch_bounds__(256) void mfvi_update_kernel(
    const float* __restrict__ Sbe, const float* __restrict__ Seb,
    const float* __restrict__ Sbb, const float* __restrict__ See,
    const float* __restrict__ Scb, const float* __restrict__ Sce,
    const uint8_t* __restrict__ em,
    const float* __restrict__ qb,  const float* __restrict__ qe,
    const float* __restrict__ qTb, const float* __restrict__ qTe,
    float* __restrict__ out_b, float* __restrict__ out_e) {

    int bh = blockIdx.x;
    int b  = bh >> 6;
    int h  = bh & 63;

    __shared__ float sh_qe[Sv];
    __shared__ float sh_qb[Sv];
    __shared__ float sh_em[Sv];
    __shared__ float part_b[4][16];   // k-half-1 partials per m-tile
    __shared__ float part_e[4][16];

    int t = threadIdx.x;
    if (t < Sv) {
        sh_qe[t] = qe[bh * Sv + t];
        sh_qb[t] = qb[bh * Sv + t];
        sh_em[t] = em[bh * Sv + t] ? 1.0f : 0.0f;
    }

    int wave  = t >> 5;
    int mtile = wave & 3;         // m-tile 0..3
    int kh    = wave >> 2;        // k-half 0..1
    int lane  = t & 31;
    int khalf = lane >> 4;        // K = 2*khalf + {0,1} within each 4-chunk
    int mr    = lane & 15;        // A row (M = lane%16)
    int m     = mtile * 16 + mr;
    int kbase = kh * 32;

    size_t rowoff = ((size_t)bh << 12) + ((size_t)m << 6);   // [b,h,m,0]
    const float* pbe = Sbe + rowoff;
    const float* peb = Seb + rowoff;
    const float* pbb = Sbb + rowoff;
    const float* pee = See + rowoff;
    const float* pcb = Scb + rowoff;
    const float* pce = Sce + rowoff;
    size_t qtoff = ((size_t)b << 12) + ((size_t)m << 6);     // qT[b,m,0]
    const float* ptb = qTb + qtoff;
    const float* pte = qTe + qtoff;

    // Kick HBM fetch of this wave's k-half of each row before the preamble.
    __builtin_prefetch(pbe + kbase, 0, 3);
    __builtin_prefetch(peb + kbase, 0, 3);
    __builtin_prefetch(pbb + kbase, 0, 3);
    __builtin_prefetch(pee + kbase, 0, 3);
    __builtin_prefetch(pcb + kbase, 0, 3);
    __builtin_prefetch(pce + kbase, 0, 3);

    __syncthreads();

    float em_m = sh_em[m];
    float hnem = (h != m) ? 1.0f : 0.0f;   // Seb mask carries (h != m)

    v8f accb = {0.f, 0.f, 0.f, 0.f, 0.f, 0.f, 0.f, 0.f};
    v8f acce = {0.f, 0.f, 0.f, 0.f, 0.f, 0.f, 0.f, 0.f};

    // Software-pipelined: step-0 loads up front, prefetch ks+1 before use.
    int kc = kbase + 2 * khalf;
    v2f vbe = *(const v2f*)(pbe + kc);
    v2f veb = *(const v2f*)(peb + kc);
    v2f vbb = *(const v2f*)(pbb + kc);
    v2f vee = *(const v2f*)(pee + kc);
    v2f vcb = *(const v2f*)(pcb + kc);
    v2f vce = *(const v2f*)(pce + kc);
    v2f vtb = *(const v2f*)(ptb + kc);
    v2f vte = *(const v2f*)(pte + kc);

#pragma unroll
    for (int ks = 0; ks < 8; ++ks) {
        int k0 = kbase + 4 * ks + 2 * khalf;
        int k1 = k0 + 1;

        // Issue next-step loads first (independent of current consumers).
        v2f nbe = vbe, neb = veb, nbb = vbb, nee = vee,
            ncb = vcb, nce = vce, ntb = vtb, nte = vte;
        if (ks < 7) {
            int kn = k0 + 4;
            nbe = *(const v2f*)(pbe + kn);
            neb = *(const v2f*)(peb + kn);
            nbb = *(const v2f*)(pbb + kn);
            nee = *(const v2f*)(pee + kn);
            ncb = *(const v2f*)(pcb + kn);
            nce = *(const v2f*)(pce + kn);
            ntb = *(const v2f*)(ptb + kn);
            nte = *(const v2f*)(pte + kn);
        }

        float emk0 = sh_em[k0], emk1 = sh_em[k1];
        float sib0 = em_m * emk0 * ((h != k0) ? 1.0f : 0.0f);
        float sib1 = em_m * emk1 * ((h != k1) ? 1.0f : 0.0f);
        float m2o0 = sib0 * ((m != k0) ? 1.0f : 0.0f);
        float m2o1 = sib1 * ((m != k1) ? 1.0f : 0.0f);
        float seb0 = em_m * emk0 * hnem;
        float seb1 = em_m * emk1 * hnem;

        v2f a_be = { vbe.x * sib0,         vbe.y * sib1 };
        v2f a_bb = { vbb.x * m2o0,         vbb.y * m2o1 };
        v2f a_cb = { vcb.x * m2o0 * vtb.x, vcb.y * m2o1 * vtb.y };
        v2f a_eb = { veb.x * seb0,         veb.y * seb1 };
        v2f a_ee = { vee.x * m2o0,         vee.y * m2o1 };
        v2f a_ce = { vce.x * m2o0 * vte.x, vce.y * m2o1 * vte.y };

        // Replicated-vector B (per-lane K mapping matches A's).
        v2f b_qe  = { sh_qe[k0], sh_qe[k1] };
        v2f b_qb  = { sh_qb[k0], sh_qb[k1] };
        v2f b_one = { 1.0f, 1.0f };

        accb = __builtin_amdgcn_wmma_f32_16x16x4_f32(false, a_be, false, b_qe,
                                                     (short)0, accb, false, false);
        accb = __builtin_amdgcn_wmma_f32_16x16x4_f32(false, a_bb, false, b_qb,
                                                     (short)0, accb, false, false);
        accb = __builtin_amdgcn_wmma_f32_16x16x4_f32(false, a_cb, false, b_one,
                                                     (short)0, accb, false, false);
        acce = __builtin_amdgcn_wmma_f32_16x16x4_f32(false, a_eb, false, b_qb,
                                                     (short)0, acce, false, false);
        acce = __builtin_amdgcn_wmma_f32_16x16x4_f32(false, a_ee, false, b_qe,
                                                     (short)0, acce, false, false);
        acce = __builtin_amdgcn_wmma_f32_16x16x4_f32(false, a_ce, false, b_one,
                                                     (short)0, acce, false, false);

        vbe = nbe; veb = neb; vbb = nbb; vee = nee;
        vcb = ncb; vce = nce; vtb = ntb; vte = nte;
    }

    // Column N=0 of D: lane 0 -> M=tile+0..7 in acc[0..7],
    // lane 16 -> M=tile+8..15 in acc[0..7].
    if (kh == 1) {                     // stash k-half-1 partials in LDS
        if (lane == 0) {
#pragma unroll
            for (int j = 0; j < 8; ++j) {
                part_b[mtile][j] = accb[j];
                part_e[mtile][j] = acce[j];
            }
        } else if (lane == 16) {
#pragma unroll
            for (int j = 0; j < 8; ++j) {
                part_b[mtile][8 + j] = accb[j];
                part_e[mtile][8 + j] = acce[j];
            }
        }
    }
    __syncthreads();

    if (kh == 0) {                     // combine (fixed order) and store
        if (lane == 0) {
#pragma unroll
            for (int j = 0; j < 8; ++j) {
                size_t o = (size_t)bh * Sv + mtile * 16 + j;
                out_b[o] += accb[j] + part_b[mtile][j];
                out_e[o] += acce[j] + part_e[mtile][j];
            }
        } else if (lane == 16) {
#pragma unroll
            for (int j = 0; j < 8; ++j) {
                size_t o = (size_t)bh * Sv + mtile * 16 + 8 + j;
                out_b[o] += accb[j] + part_b[mtile][8 + j];
                out_e[o] += acce[j] + part_e[mtile][8 + j];
            }
        }
    }
}

// ---------------------------------------------------------------------------
extern "C" void kernel_launch(void* const* d_in, const int* in_sizes, int n_in,
                              void* d_out, int out_size, void* d_ws, size_t ws_size,
                              hipStream_t stream) {
    const float*   s_arg_begin = (const float*)d_in[0];
    const float*   s_arg_end   = (const float*)d_in[1];
    const float*   Sbe         = (const float*)d_in[2];
    const float*   Seb         = (const float*)d_in[3];
    const float*   Sbb         = (const float*)d_in[4];
    const float*   See         = (const float*)d_in[5];
    const float*   Scb         = (const float*)d_in[6];
    const float*   Sce         = (const float*)d_in[7];
    const uint8_t* em          = (const uint8_t*)d_in[8];  // numpy bool, 1B/elem

    float* out_b = (float*)d_out;
    float* out_e = out_b + BSS;

    float* w   = (float*)d_ws;             // 4 * 64KB scratch
    float* qb  = w;
    float* qe  = w + BSS;
    float* qTb = w + 2 * BSS;
    float* qTe = w + 3 * BSS;

    hipMemcpyAsync(out_b, s_arg_begin, BSS * sizeof(float),
                   hipMemcpyDeviceToDevice, stream);
    hipMemcpyAsync(out_e, s_arg_end, BSS * sizeof(float),
                   hipMemcpyDeviceToDevice, stream);

    for (int it = 0; it < 3; ++it) {
        mfvi_sigmoid_kernel<<<BSS / 256, 256, 0, stream>>>(out_b, out_e,
                                                           qb, qe, qTb, qTe);
        mfvi_update_kernel<<<BH, 256, 0, stream>>>(Sbe, Seb, Sbb, See, Scb, Sce,
                                                   em, qb, qe, qTb, qTe,
                                                   out_b, out_e);
    }
}